// SparseLinear_64965675319395
// MI455X (gfx1250) — compile-verified
//
#include <hip/hip_runtime.h>
#include <stdint.h>

// SparseLinear: out[b,g,o] = sum_k x_gathered[b,g,k] * W[g,o,k] + bias[g,o]
//   B=256, N_IN=128, F_IN=64, G=128, F_OUT=64, S=16, K = S*F_IN = 1024
// fp32 WMMA (v_wmma_f32_16x16x4_f32), staged via GLOBAL_LOAD_ASYNC_TO_LDS_B128
// (no VGPR staging -> no scratch spills), double-buffered LDS, ASYNCcnt sync.

typedef __attribute__((ext_vector_type(2))) float v2f;
typedef __attribute__((ext_vector_type(8))) float v8f;

#define LDS_STRIDE 68   // 64 floats + 4 pad: float4-aligned rows, banks spread

__global__ __launch_bounds__(256) void sparse_linear_wmma_f32(
    const float* __restrict__ x,     // (256,128,64)
    const int*   __restrict__ idx,   // (128,16)
    const float* __restrict__ W,     // (128,64,1024)
    const float* __restrict__ bias,  // (128,64)
    float*       __restrict__ out)   // (256,128,64)
{
    __shared__ float As[2][32 * LDS_STRIDE];   // gathered x panel: 32 rows x 64
    __shared__ float Ws[2][64 * LDS_STRIDE];   // W panel: 64 rows (o) x 64 (k-chunk)

    const int g    = blockIdx.y;        // group 0..127
    const int b0   = blockIdx.x * 32;   // batch tile base
    const int tid  = threadIdx.x;
    const int lane = tid & 31;          // wave32
    const int wave = tid >> 5;          // 0..7
    const int mt   = wave >> 2;         // 0..1 : 16-row batch subtile
    const int nt   = wave & 3;          // 0..3 : 16-col output subtile

    // ---- per-thread async-copy slots (b128 per issue, 6 issues per K-chunk) ----
    // A chunk: 32 rows x 64 f32 = 512 float4 -> 2 per thread
    // W chunk: 64 rows x 64 f32 = 1024 float4 -> 4 per thread
    unsigned ldsA[2], ldsW[4];   // LDS byte offsets (buffer 0)
    uint64_t gA[2],   gW[4];     // global base addresses (chunk-invariant part)
    #pragma unroll
    for (int i = 0; i < 2; ++i) {
        int e = tid + i * 256, r = e >> 4, c4 = e & 15;
        ldsA[i] = (unsigned)(uintptr_t)&As[0][r * LDS_STRIDE + c4 * 4];
        // x[b0+r, nrow, c4*4..]: base without the gather row (nrow added per chunk)
        gA[i]   = (uint64_t)(uintptr_t)(x + (size_t)(b0 + r) * (128 * 64) + c4 * 4);
    }
    #pragma unroll
    for (int i = 0; i < 4; ++i) {
        int e = tid + i * 256, r = e >> 4, c4 = e & 15;
        ldsW[i] = (unsigned)(uintptr_t)&Ws[0][r * LDS_STRIDE + c4 * 4];
        gW[i]   = (uint64_t)(uintptr_t)(W + (size_t)(g * 64 + r) * 1024 + c4 * 4);
    }
    const unsigned aBufB = (unsigned)sizeof(As[0]);   // bytes per A buffer
    const unsigned wBufB = (unsigned)sizeof(Ws[0]);   // bytes per W buffer

    auto stage_async = [&](int s, int p) {
        const uint64_t aoff = (uint64_t)idx[g * 16 + s] * (64 * 4); // gather row bytes
        const uint64_t woff = (uint64_t)s * (64 * 4);               // k-chunk bytes
        #pragma unroll
        for (int i = 0; i < 2; ++i) {
            unsigned l = ldsA[i] + (unsigned)p * aBufB;
            uint64_t a = gA[i] + aoff;
            asm volatile("global_load_async_to_lds_b128 %0, %1, off"
                         :: "v"(l), "v"(a) : "memory");
        }
        #pragma unroll
        for (int i = 0; i < 4; ++i) {
            unsigned l = ldsW[i] + (unsigned)p * wBufB;
            uint64_t a = gW[i] + woff;
            asm volatile("global_load_async_to_lds_b128 %0, %1, off"
                         :: "v"(l), "v"(a) : "memory");
        }
    };

    v8f acc0 = {};
    v8f acc1 = {};

    // A fragment (16x4 f32): lane = M (l&15); v0: K=k0+0 / +2 per lane half; v1: +1 / +3
    const int aoff = (mt * 16 + (lane & 15)) * LDS_STRIDE + ((lane >> 4) * 2);
    // B fragment (4x16 f32): B[k][n] = W[o=n][k]; lane = N, same K striping.
    const int woff = (nt * 16 + (lane & 15)) * LDS_STRIDE + ((lane >> 4) * 2);

    // Prologue: stage chunk 0 into buffer 0.
    stage_async(0, 0);
    asm volatile("s_wait_asynccnt 0x0" ::: "memory");
    __syncthreads();

    int p = 0;
    for (int s = 0; s < 16; ++s) {
        if (s + 1 < 16) stage_async(s + 1, p ^ 1);   // async fill of other buffer

        const float* Ab = &As[p][aoff];
        const float* Wb = &Ws[p][woff];
        #pragma unroll
        for (int k0 = 0; k0 < 64; k0 += 8) {         // 2 WMMAs/step, 2 acc chains
            v2f a0 = *(const v2f*)(Ab + k0);
            v2f w0 = *(const v2f*)(Wb + k0);
            acc0 = __builtin_amdgcn_wmma_f32_16x16x4_f32(
                false, a0, false, w0, (short)0, acc0, false, false);
            v2f a1 = *(const v2f*)(Ab + k0 + 4);
            v2f w1 = *(const v2f*)(Wb + k0 + 4);
            acc1 = __builtin_amdgcn_wmma_f32_16x16x4_f32(
                false, a1, false, w1, (short)0, acc1, false, false);
        }

        if (s + 1 < 16) {
            asm volatile("s_wait_asynccnt 0x0" ::: "memory");
            __syncthreads();
            p ^= 1;
        }
    }

    v8f acc = acc0 + acc1;

    // Epilogue: C/D layout — VGPR v, lanes 0-15 => M=v, lanes 16-31 => M=v+8.
    const int o  = nt * 16 + (lane & 15);
    const float bv = bias[g * 64 + o];
    #pragma unroll
    for (int v = 0; v < 8; ++v) {
        int m = mt * 16 + v + ((lane >> 4) * 8);
        out[((size_t)(b0 + m) * 128 + g) * 64 + o] = acc[v] + bv;
    }
}

extern "C" void kernel_launch(void* const* d_in, const int* in_sizes, int n_in,
                              void* d_out, int out_size, void* d_ws, size_t ws_size,
                              hipStream_t stream) {
    const float* x    = (const float*)d_in[0];
    const int*   idx  = (const int*)  d_in[1];
    const float* W    = (const float*)d_in[2];
    const float* bias = (const float*)d_in[3];
    float*       out  = (float*)d_out;

    dim3 grid(8, 128);   // (B/32, G)
    sparse_linear_wmma_f32<<<grid, 256, 0, stream>>>(x, idx, W, bias, out);
}